// Varifold_loss_11596411699384
// MI455X (gfx1250) — compile-verified
//
#include <hip/hip_runtime.h>
#include <hip/hip_bf16.h>

typedef float v2f __attribute__((ext_vector_type(2)));
typedef float v8f __attribute__((ext_vector_type(8)));

#define LOG2E 1.4426950408889634f

// ---------------------------------------------------------------------------
// Phase 1: per-face quantities for 4 meshes (2 samples x {pred, targ}).
// Writes, per mesh:
//   packedC [G][64] : WMMA operand layout, lane l<16 -> (Cx,Cy), l>=16 -> (Cz,0)
//   packedN [G][64] : same layout for unit normals
//   R       [Fn]    : |C|^2
//   L       [Fn]    : face area (0.5*|cross|, with 1e-24 eps under sqrt)
// ---------------------------------------------------------------------------
__global__ void mesh_quant_kernel(const float* __restrict__ pred,
                                  const float* __restrict__ targ,
                                  const int* __restrict__ faces,
                                  float* __restrict__ ws,
                                  int Vn, int Fn, int G, int MS) {
    int t = blockIdx.x * blockDim.x + threadIdx.x;
    if (t >= 4 * Fn) return;
    int mesh = t / Fn;
    int f    = t - mesh * Fn;
    int s    = mesh >> 1;
    const float* V = ((mesh & 1) ? targ : pred) + (size_t)s * Vn * 3;

    int i0 = faces[f * 3 + 0], i1 = faces[f * 3 + 1], i2 = faces[f * 3 + 2];
    float x0 = V[i0 * 3 + 0], y0 = V[i0 * 3 + 1], z0 = V[i0 * 3 + 2];
    float x1 = V[i1 * 3 + 0], y1 = V[i1 * 3 + 1], z1 = V[i1 * 3 + 2];
    float x2 = V[i2 * 3 + 0], y2 = V[i2 * 3 + 1], z2 = V[i2 * 3 + 2];

    const float third = 1.0f / 3.0f;
    float cx = (x0 + x1 + x2) * third;
    float cy = (y0 + y1 + y2) * third;
    float cz = (z0 + z1 + z2) * third;

    float e1x = x1 - x0, e1y = y1 - y0, e1z = z1 - z0;
    float e2x = x2 - x0, e2y = y2 - y0, e2z = z2 - z0;
    float nx = 0.5f * (e1y * e2z - e1z * e2y);
    float ny = 0.5f * (e1z * e2x - e1x * e2z);
    float nz = 0.5f * (e1x * e2y - e1y * e2x);
    float len = sqrtf(nx * nx + ny * ny + nz * nz + 1e-24f);
    float inv = 1.0f / len;
    nx *= inv; ny *= inv; nz *= inv;

    float* mb = ws + (size_t)mesh * MS;
    float* pc = mb;
    float* pn = mb + (size_t)G * 64;
    float* pr = mb + (size_t)2 * G * 64;
    float* pl = pr + Fn;

    int g = f >> 4, sl = f & 15;
    float* c0 = pc + (size_t)g * 64;
    c0[sl * 2 + 0]      = cx;
    c0[sl * 2 + 1]      = cy;
    c0[32 + sl * 2 + 0] = cz;
    c0[32 + sl * 2 + 1] = 0.0f;
    float* n0 = pn + (size_t)g * 64;
    n0[sl * 2 + 0]      = nx;
    n0[sl * 2 + 1]      = ny;
    n0[32 + sl * 2 + 0] = nz;
    n0[32 + sl * 2 + 1] = 0.0f;

    pr[f] = cx * cx + cy * cy + cz * cz;
    pl[f] = len;
}

// ---------------------------------------------------------------------------
// Per-row tile loop: accumulates this lane's share of
//   sum_{j in columns} L1[i] * K(i,j) * L2[j]
// over tiles gj = gj0, gj0+gjstep, ... < G.
// If sym==true, off-diagonal tiles (gj != gi) are weighted x2 (upper
// triangle only is visited by the caller).
// ---------------------------------------------------------------------------
__device__ __forceinline__ float tile_row_accum(
        const float* __restrict__ pc1, const float* __restrict__ pn1,
        const float* __restrict__ pr1, const float* __restrict__ pl1,
        const float* __restrict__ pc2, const float* __restrict__ pn2,
        const float* __restrict__ pr2, const float* __restrict__ pl2,
        int gi, int gj0, int gjstep, int G, int lane, bool sym) {

    // A-matrix operands (16x4 f32: 2 VGPRs/lane), fixed for the whole row
    v2f aC = *(const v2f*)(pc1 + (size_t)gi * 64 + lane * 2);
    v2f aN = *(const v2f*)(pn1 + (size_t)gi * 64 + lane * 2);

    // This lane's 8 row elements: m = r + (lane>=16 ? 8 : 0)
    int mo = gi * 16 + ((lane >= 16) ? 8 : 0);
    float l1v[8], r1v[8];
#pragma unroll
    for (int r = 0; r < 8; ++r) { l1v[r] = pl1[mo + r]; r1v[r] = pr1[mo + r]; }

    // exp2-domain constants: exponent * log2(e)
    const float is2[3] = { 2500.0f * LOG2E, 10000.0f * LOG2E, 15625.0f * LOG2E };
    const float wsg[3] = { 1.0f, 0.25f, 0.16f };   // (sig/sig0)^2

    float acc = 0.0f;
    for (int gj = gj0; gj < G; gj += gjstep) {
        v2f bC = *(const v2f*)(pc2 + (size_t)gj * 64 + lane * 2);
        v2f bN = *(const v2f*)(pn2 + (size_t)gj * 64 + lane * 2);

        v8f cd = {0.f,0.f,0.f,0.f,0.f,0.f,0.f,0.f};
        v8f nd = {0.f,0.f,0.f,0.f,0.f,0.f,0.f,0.f};
        cd = __builtin_amdgcn_wmma_f32_16x16x4_f32(false, aC, false, bC,
                                                   (short)0, cd, false, false);
        nd = __builtin_amdgcn_wmma_f32_16x16x4_f32(false, aN, false, bN,
                                                   (short)0, nd, false, false);

        int   jn  = gj * 16 + (lane & 15);  // this lane's column index
        float r2  = pr2[jn];
        float l2w = pl2[jn];
        if (sym && gj != gi) l2w *= 2.0f;   // count (i,j) and (j,i)

#pragma unroll
        for (int r = 0; r < 8; ++r) {
            float negd2 = 2.0f * cd[r] - r1v[r] - r2;                // -d2
            float basen = fmaf(nd[r], 4.0f * LOG2E, -4.0f * LOG2E); // (4*ndot-4)*log2e
            float e = 0.0f;
#pragma unroll
            for (int k = 0; k < 3; ++k)
                e = fmaf(wsg[k],
                         __builtin_amdgcn_exp2f(fmaf(negd2, is2[k], basen)),
                         e);
            acc = fmaf(l1v[r] * l2w, e, acc);
        }
    }
    return acc;
}

// ---------------------------------------------------------------------------
// Phase 2: wave jobs.
//  Jobs [0, NJOBS_SYM): symmetric products pss/ptt, upper triangle only,
//    each wave handles the row pair (ri, G-1-ri) for constant work (~G+1 tiles
//    per NCHUNK group), columns strided by NCHUNK.
//  Jobs [NJOBS_SYM, NJOBS): pst (pred x targ), full rows, weight -2.
// ---------------------------------------------------------------------------
__global__ void varifold_main_kernel(const float* __restrict__ ws,
                                     float* __restrict__ partials,
                                     int Fn, int G, int MS,
                                     int NCHUNK, int NJOBS_SYM) {
    int gid  = blockIdx.x * blockDim.x + threadIdx.x;
    int wave = gid >> 5;
    int lane = gid & 31;

    float acc, wprod;
    if (wave < NJOBS_SYM) {
        int HG    = G >> 1;
        int chunk = wave % NCHUNK;
        int w2    = wave / NCHUNK;
        int ri    = w2 % HG;  w2 /= HG;
        int ps    = w2 % 2;            // 0 -> pss (pred), 1 -> ptt (targ)
        int s     = w2 / 2;            // sample 0/1
        int m     = s * 2 + ps;

        const float* b  = ws + (size_t)m * MS;
        const float* pc = b;
        const float* pn = b + (size_t)G * 64;
        const float* pr = b + (size_t)2 * G * 64;
        const float* pl = pr + Fn;

        int gi1 = ri, gi2 = G - 1 - ri;
        acc  = tile_row_accum(pc, pn, pr, pl, pc, pn, pr, pl,
                              gi1, gi1 + chunk, NCHUNK, G, lane, true);
        acc += tile_row_accum(pc, pn, pr, pl, pc, pn, pr, pl,
                              gi2, gi2 + chunk, NCHUNK, G, lane, true);
        wprod = 1.0f;
    } else {
        int w     = wave - NJOBS_SYM;
        int chunk = w % NCHUNK;
        int w2    = w / NCHUNK;
        int gi    = w2 % G;
        int s     = w2 / G;
        int m1    = s * 2 + 0;         // pred
        int m2    = s * 2 + 1;         // targ

        const float* b1  = ws + (size_t)m1 * MS;
        const float* b2  = ws + (size_t)m2 * MS;
        acc = tile_row_accum(b1, b1 + (size_t)G * 64,
                             b1 + (size_t)2 * G * 64, b1 + (size_t)2 * G * 64 + Fn,
                             b2, b2 + (size_t)G * 64,
                             b2 + (size_t)2 * G * 64, b2 + (size_t)2 * G * 64 + Fn,
                             gi, chunk, NCHUNK, G, lane, false);
        wprod = -2.0f;
    }

    // wave32 reduction
#pragma unroll
    for (int off = 16; off > 0; off >>= 1)
        acc += __shfl_xor(acc, off, 32);

    if (lane == 0) partials[wave] = wprod * acc;
}

// ---------------------------------------------------------------------------
// Phase 3: deterministic fixed-order reduction of wave partials.
// ---------------------------------------------------------------------------
__global__ void varifold_reduce_kernel(const float* __restrict__ partials,
                                       int n, float* __restrict__ out) {
    __shared__ float sm[256];
    float a = 0.0f;
    for (int i = threadIdx.x; i < n; i += 256) a += partials[i];
    sm[threadIdx.x] = a;
    __syncthreads();
    for (int sft = 128; sft > 0; sft >>= 1) {
        if ((int)threadIdx.x < sft) sm[threadIdx.x] += sm[threadIdx.x + sft];
        __syncthreads();
    }
    if (threadIdx.x == 0) out[0] = 0.5f * sm[0];   // mean over B=2
}

extern "C" void kernel_launch(void* const* d_in, const int* in_sizes, int n_in,
                              void* d_out, int out_size, void* d_ws, size_t ws_size,
                              hipStream_t stream) {
    const float* pred  = (const float*)d_in[0];
    const float* targ  = (const float*)d_in[1];
    const int*   faces = (const int*)d_in[2];

    const int B  = 2;
    const int Fn = in_sizes[2] / 3;        // 6144
    const int Vn = in_sizes[0] / (B * 3);  // 5000
    const int G  = Fn / 16;                // 384 tile groups
    const int MS = 2 * G * 64 + 2 * Fn;    // floats per mesh in workspace

    float* ws       = (float*)d_ws;
    float* partials = ws + (size_t)4 * MS;

    const int NCHUNK    = 4;
    const int NJOBS_SYM = B * 2 * (G / 2) * NCHUNK;  // 3072 waves, row pairs
    const int NJOBS_PST = B * G * NCHUNK;            // 3072 waves
    const int NJOBS     = NJOBS_SYM + NJOBS_PST;     // 6144 waves

    mesh_quant_kernel<<<(4 * Fn + 255) / 256, 256, 0, stream>>>(
        pred, targ, faces, ws, Vn, Fn, G, MS);

    varifold_main_kernel<<<(NJOBS * 32 + 255) / 256, 256, 0, stream>>>(
        ws, partials, Fn, G, MS, NCHUNK, NJOBS_SYM);

    varifold_reduce_kernel<<<1, 256, 0, stream>>>(partials, NJOBS, (float*)d_out);
}